// CombinedOrthogonalAdapter_15015205667206
// MI455X (gfx1250) — compile-verified
//
#include <hip/hip_runtime.h>

typedef __attribute__((ext_vector_type(2))) float v2f;
typedef __attribute__((ext_vector_type(4))) float v4f;
typedef __attribute__((ext_vector_type(8))) float v8f;
typedef __attribute__((ext_vector_type(4))) unsigned int u32x4;
typedef __attribute__((ext_vector_type(4))) int i32x4;
typedef __attribute__((ext_vector_type(8))) int i32x8;

constexpr int E_    = 8;
constexpr int R_    = 64;
constexpr int D_    = 2048;
constexpr int NTOK  = 16384;   // B*S
constexpr int TOK   = 16;      // tokens per tile (WMMA M)
constexpr int BK    = 128;     // K chunk staged in LDS
constexpr int XS_ST = BK + 4;  // TDM pad: 4 DWORDs every 128 DWORDs
constexpr int HS_ST = R_ + 4;
constexpr int NCHUNK = D_ / BK;

// ---------------------------------------------------------------- bucketing
__global__ void zero_counts_kernel(int* counts) {
  if (threadIdx.x < E_) counts[threadIdx.x] = 0;
}

__global__ void bucket_kernel(const int* __restrict__ task_idx,
                              int* __restrict__ counts,
                              int* __restrict__ lists) {
  int t = blockIdx.x * blockDim.x + threadIdx.x;
  if (t < NTOK) {
    int e = task_idx[t];
    int pos = atomicAdd(&counts[e], 1);
    lists[e * NTOK + pos] = t;
  }
}

// --------------------------------------------------- TDM gather of 16 x-rows
// D# per CDNA5 ISA ch.8: group0 = {flags, lds_addr, global_addr, type},
// group1 = {pad/data_size, tensor dims, tile dims, stride}, groups 2/3 =
// 16-bit gather row indices (token ids). One instruction moves the whole
// 16 x 128 f32 tile (scattered rows) into LDS with the padded stride.
__device__ __forceinline__ void tdm_gather_x(unsigned long long gaddr,
                                             unsigned int laddr,
                                             i32x4 idxlo, i32x4 idxhi) {
  u32x4 g0;
  g0[0] = 0x80000001u;                               // count=1 | gather_mode, 16-bit idx
  g0[1] = laddr;                                     // lds_addr (bytes)
  g0[2] = (unsigned int)(gaddr & 0xffffffffull);     // global_addr[31:0]
  g0[3] = (unsigned int)((gaddr >> 32) & 0x01ffffffull) | 0x80000000u; // [56:32] | type=2
  i32x8 g1;
  g1[0] = (2 << 16)                                  // data_size = 4 bytes
        | (1 << 20)                                  // pad_enable
        | (6 << 22)                                  // pad_interval: every 128 DWORDs
        | (3 << 25);                                 // pad_amount: 4 DWORDs
  g1[1] = (int)((D_ & 0xffff) << 16);                // tensor_dim0[15:0]   -> bits 63:48
  g1[2] = (int)((D_ >> 16) | ((NTOK & 0xffff) << 16));   // dim0 hi | dim1 lo
  g1[3] = (int)((NTOK >> 16) | (BK << 16));          // dim1 hi | tile_dim0 = 128
  g1[4] = TOK;                                       // tile_dim1 = #gather indices = 16
  g1[5] = D_;                                        // tensor_dim0_stride = 2048
  g1[6] = 0;
  g1[7] = 0;
  i32x8 gz = {};                                     // unused trailing group (zero)
  __builtin_amdgcn_tensor_load_to_lds(g0, g1, idxlo, idxhi, gz, 0);
}

// ------------------------------------------------------- grouped LoRA GEMM
__launch_bounds__(256)
__global__ void lora_tile_kernel(const float* __restrict__ x,
                                 const float* __restrict__ lora_A,   // [E,R,D]
                                 const float* __restrict__ lora_B,   // [E,D,R]
                                 const int*   __restrict__ counts,
                                 const int*   __restrict__ lists,
                                 float* __restrict__ out) {
  __shared__ float xs[2][TOK * XS_ST];     // double-buffered TDM x tile (16.9 KB)
  __shared__ float hp[8 * TOK * R_];       // per-wave H partials       (32 KB)
  __shared__ float Hs[TOK * HS_ST];        // reduced H                 (4.4 KB)
  __shared__ int   tids[TOK];

  const int e    = blockIdx.y;
  const int tile = blockIdx.x;
  const int cnt  = counts[e];
  if (tile * TOK >= cnt) return;           // uniform early-out

  const int tid  = threadIdx.x;
  const int wave = tid >> 5;               // 0..7
  const int lane = tid & 31;
  const int lr   = lane & 15;              // N / M index within half-wave
  const int lh   = lane >> 4;              // half selector

  if (tid < TOK) {
    int c = tile * TOK + tid;
    tids[tid] = (c < cnt) ? lists[e * NTOK + c] : -1;
  }
  __syncthreads();

  // pack clamped token ids as 16-bit TDM gather indices (uniform values)
  i32x4 idxlo, idxhi;
  #pragma unroll
  for (int i = 0; i < 4; ++i) {
    int a0 = tids[2 * i];     a0 = a0 < 0 ? 0 : a0;
    int a1 = tids[2 * i + 1]; a1 = a1 < 0 ? 0 : a1;
    int b0 = tids[8 + 2 * i];     b0 = b0 < 0 ? 0 : b0;
    int b1 = tids[8 + 2 * i + 1]; b1 = b1 < 0 ? 0 : b1;
    idxlo[i] = (a0 & 0xffff) | (a1 << 16);
    idxhi[i] = (b0 & 0xffff) | (b1 << 16);
  }

  // Phase-1 accumulators: H[16x64] as four 16x16 N-tiles
  v8f acc0 = {}, acc1 = {}, acc2 = {}, acc3 = {};

  // per-lane base into lora_A: B-operand b = A^T[k][r] = lora_A[e][r][k]
  const float* aw = lora_A + ((size_t)e * R_ + lr) * D_ + 2 * lh;

  // warm L2 for the phase-2 weight rows this wave will consume
  #pragma unroll
  for (int nt = 0; nt < 16; ++nt) {
    __builtin_prefetch(lora_B + ((size_t)e * D_ + wave * 256 + nt * 16) * R_, 0, 1);
  }

  // prologue: TDM-issue chunk 0 into buffer 0 (one wave drives the DMA)
  if (wave == 0) {
    tdm_gather_x((unsigned long long)(uintptr_t)x,
                 (unsigned int)(uintptr_t)&xs[0][0], idxlo, idxhi);
  }

  // ---------------- Phase 1: H = X(16xD) @ A_e^T(DxR), K split over 8 waves
  for (int ch = 0; ch < NCHUNK; ++ch) {
    const int cur = ch & 1;
    if (wave == 0) __builtin_amdgcn_s_wait_tensorcnt(0);
    __syncthreads();                       // xs[cur] ready for all waves;
                                           // previous consumers of xs[1-cur] drained
    if (wave == 0 && (ch + 1) < NCHUNK) {
      tdm_gather_x((unsigned long long)(uintptr_t)(x + (ch + 1) * BK),
                   (unsigned int)(uintptr_t)&xs[1 - cur][0], idxlo, idxhi);
    }

    const float* xcur = &xs[cur][0];
    const int kbase = wave * 16;           // this wave's K sub-slice of the chunk
    #pragma unroll
    for (int ks = 0; ks < 4; ++ks) {
      const int kk = kbase + ks * 4;
      // A operand: 16x4 tile of x from LDS (lane: row lr, k = kk + 2*lh)
      v2f a = *(const v2f*)&xcur[lr * XS_ST + kk + 2 * lh];
      const float* awk = aw + (size_t)(ch * BK + kk);
      v2f b0 = *(const v2f*)(awk + (size_t)0  * D_);
      v2f b1 = *(const v2f*)(awk + (size_t)16 * D_);
      v2f b2 = *(const v2f*)(awk + (size_t)32 * D_);
      v2f b3 = *(const v2f*)(awk + (size_t)48 * D_);
      acc0 = __builtin_amdgcn_wmma_f32_16x16x4_f32(false, a, false, b0, (short)0, acc0, false, false);
      acc1 = __builtin_amdgcn_wmma_f32_16x16x4_f32(false, a, false, b1, (short)0, acc1, false, false);
      acc2 = __builtin_amdgcn_wmma_f32_16x16x4_f32(false, a, false, b2, (short)0, acc2, false, false);
      acc3 = __builtin_amdgcn_wmma_f32_16x16x4_f32(false, a, false, b3, (short)0, acc3, false, false);
    }
    __syncthreads();                       // done reading xs[cur] before it is
                                           // re-filled by the next iteration's TDM issue
  }

  // spill per-wave partials; C/D layout: elem(v,lane) = H[v + 8*lh][nt*16 + lr]
  {
    float* hpw = &hp[wave * (TOK * R_)];
    #pragma unroll
    for (int v = 0; v < 8; ++v) {
      const int m = v + 8 * lh;
      hpw[m * R_ +  0 + lr] = acc0[v];
      hpw[m * R_ + 16 + lr] = acc1[v];
      hpw[m * R_ + 32 + lr] = acc2[v];
      hpw[m * R_ + 48 + lr] = acc3[v];
    }
  }
  __syncthreads();

  // tree-reduce 8 partials -> Hs (padded stride for conflict-free phase-2 reads)
  for (int i = tid; i < TOK * R_; i += 256) {
    float ssum = 0.f;
    #pragma unroll
    for (int w = 0; w < 8; ++w) ssum += hp[w * (TOK * R_) + i];
    Hs[(i >> 6) * HS_ST + (i & 63)] = ssum;
  }
  __syncthreads();

  // ---------------- Phase 2: out_tile(16xD) = H(16x64) @ B_e^T(64xD)
  const int d0w = wave * 256;              // 256 out-dims per wave
  #pragma unroll 1
  for (int nt = 0; nt < 16; ++nt) {
    const int d0 = d0w + nt * 16;
    v8f acc = {};
    // B operand: b = B^T[k][d] = lora_B[e][d][k] ; lane row d0+lr, k = 2*lh
    const float* bw = lora_B + ((size_t)e * D_ + d0 + lr) * R_ + 2 * lh;
    #pragma unroll
    for (int ks = 0; ks < 16; ++ks) {
      v2f a = *(const v2f*)&Hs[lr * HS_ST + ks * 4 + 2 * lh];
      v2f b = *(const v2f*)(bw + ks * 4);
      acc = __builtin_amdgcn_wmma_f32_16x16x4_f32(false, a, false, b, (short)0, acc, false, false);
    }
    #pragma unroll
    for (int v = 0; v < 8; ++v) {
      const int m  = v + 8 * lh;
      const int t2 = tids[m];
      if (t2 >= 0) out[(size_t)t2 * D_ + d0 + lr] = acc[v];
    }
  }
}

// ------------------------------------------------------------------ launch
extern "C" void kernel_launch(void* const* d_in, const int* in_sizes, int n_in,
                              void* d_out, int out_size, void* d_ws, size_t ws_size,
                              hipStream_t stream) {
  const float* x      = (const float*)d_in[0];
  const float* lora_A = (const float*)d_in[1];
  const float* lora_B = (const float*)d_in[2];
  const int*   tindex = (const int*)d_in[3];
  float* out = (float*)d_out;

  int* counts = (int*)d_ws;                       // 8 ints
  int* lists  = (int*)((char*)d_ws + 256);        // 8 * 16384 ints

  zero_counts_kernel<<<1, 32, 0, stream>>>(counts);
  bucket_kernel<<<(NTOK + 255) / 256, 256, 0, stream>>>(tindex, counts, lists);

  dim3 grid(NTOK / TOK, E_);                      // 1024 tiles x 8 experts
  lora_tile_kernel<<<grid, 256, 0, stream>>>(x, lora_A, lora_B, counts, lists, out);
}